// GCN_ZEncoder_21887153340937
// MI455X (gfx1250) — compile-verified
//
#include <hip/hip_runtime.h>
#include <hip/hip_bf16.h>

#define NN 8192
#define GG 64
#define HH 128
#define LN_EPS 1e-5f

typedef float v2f __attribute__((ext_vector_type(2)));
typedef float v8f __attribute__((ext_vector_type(8)));

// ---------------------------------------------------------------------------
// CDNA5 async copy: 16 bytes per lane, global -> LDS, tracked by ASYNCcnt.
// ---------------------------------------------------------------------------
__device__ __forceinline__ void async_cp16(const float* __restrict__ g, float* l) {
    const unsigned int lds_off = (unsigned int)(size_t)l;   // low 32 bits = LDS offset
    asm volatile("global_load_async_to_lds_b128 %0, %1, off"
                 :: "v"(lds_off), "v"(g)
                 : "memory");
}
__device__ __forceinline__ void wait_async0() {
    asm volatile("s_wait_asynccnt 0x0" ::: "memory");
}

// ---------------------------------------------------------------------------
// 1) lin1: x0[n,h] = dot(X[n,:], lin1_w[h,:]) + lin1_b[h]
// ---------------------------------------------------------------------------
__global__ void k_lin1(const float* __restrict__ X, const float* __restrict__ w,
                       const float* __restrict__ b, float* __restrict__ x0) {
    __shared__ float xs[GG];
    const int n = blockIdx.x;
    const int h = threadIdx.x;          // 128 threads
    if (h < GG) xs[h] = X[(size_t)n * GG + h];
    __syncthreads();
    float acc = b[h];
    const float* wr = w + (size_t)h * GG;
    #pragma unroll 8
    for (int g = 0; g < GG; ++g) acc = fmaf(xs[g], wr[g], acc);
    x0[(size_t)n * HH + h] = acc;
}

// ---------------------------------------------------------------------------
// 2) deg init to 1.0 (self-loop weight)
// ---------------------------------------------------------------------------
__global__ void k_deg_init(float* __restrict__ deg) {
    deg[blockIdx.x * 256 + threadIdx.x] = 1.0f;
}

// ---------------------------------------------------------------------------
// 3) fused: copy adj -> out tail  +  column sums of masked adj (deg accum)
//    grid 512 blocks x 256 thr; block owns 16 rows; thread owns 32 columns
// ---------------------------------------------------------------------------
__global__ void k_deg_copy(const float* __restrict__ adj, float* __restrict__ outA,
                           float* __restrict__ deg) {
    const int t = threadIdx.x;
    const int r0 = blockIdx.x * 16;
    float acc[32];
    #pragma unroll
    for (int j = 0; j < 32; ++j) acc[j] = 0.0f;
    for (int r = 0; r < 16; ++r) {
        const size_t base = (size_t)(r0 + r) * NN;
        #pragma unroll 8
        for (int j = 0; j < 32; ++j) {
            const size_t idx = base + t + j * 256;
            float v = adj[idx];
            outA[idx] = v;                       // verbatim adj_A copy to output
            acc[j] += (v > 0.0f) ? v : 0.0f;     // masked column partial sum
        }
    }
    #pragma unroll
    for (int j = 0; j < 32; ++j) atomicAdd(&deg[t + j * 256], acc[j]);
}

// ---------------------------------------------------------------------------
// 4) dinv = deg>0 ? rsqrt(deg) : 0
// ---------------------------------------------------------------------------
__global__ void k_dinv(const float* __restrict__ deg, float* __restrict__ dinv) {
    int i = blockIdx.x * 256 + threadIdx.x;
    float d = deg[i];
    dinv[i] = (d > 0.0f) ? __frsqrt_rn(d) : 0.0f;
}

// ---------------------------------------------------------------------------
// 5) y[r,h] = dinv[r] * dot(x[r,:], conv_w[h,:])    (small GEMM, VALU)
// ---------------------------------------------------------------------------
__global__ void k_xw(const float* __restrict__ x, const float* __restrict__ w,
                     const float* __restrict__ dinv, float* __restrict__ y) {
    __shared__ float xs[16][HH];
    const int n0 = blockIdx.x * 16;
    const int h = threadIdx.x;
    #pragma unroll
    for (int i = 0; i < 16; ++i) xs[i][h] = x[(size_t)(n0 + i) * HH + h];
    __syncthreads();
    float acc[16];
    #pragma unroll
    for (int i = 0; i < 16; ++i) acc[i] = 0.0f;
    const float* wr = w + (size_t)h * HH;
    for (int k = 0; k < HH; ++k) {
        const float wk = wr[k];
        #pragma unroll
        for (int i = 0; i < 16; ++i) acc[i] = fmaf(wk, xs[i][k], acc[i]);
    }
    #pragma unroll
    for (int i = 0; i < 16; ++i)
        y[(size_t)(n0 + i) * HH + h] = dinv[n0 + i] * acc[i];
}

// ---------------------------------------------------------------------------
// 6) dominant GEMM via fp32 WMMA + async double-buffered LDS staging:
//    zT[h, c] = sum_r y[r, h] * max(adj[r, c], 0)     (self-loop added later)
//    A = y^T fragment (16h x 4r), B = adj fragment (4r x 16c)
//    block: 256 thr (8 waves), c-tile = 32, full H = 128, K-chunk = 32
//    LDS: 2 x (32x32 + 32x128) floats = 40 KB
// ---------------------------------------------------------------------------
#define KC 32
#define CT 32
__global__ void k_spmm(const float* __restrict__ adj, const float* __restrict__ y,
                       float* __restrict__ zT) {
    __shared__ float aS[2][KC][CT];   // raw adjacency chunks (masked on read)
    __shared__ float yS[2][KC][HH];   // y chunks

    const int tid  = threadIdx.x;
    const int lane = tid & 31;
    const int wave = tid >> 5;          // 0..7 -> m-tile (h0 = 16*wave)
    const int c0   = blockIdx.x * CT;
    const int h0   = wave * 16;
    const int m    = lane & 15;         // M (A) / N (B) index within tile
    const int kb   = (lane < 16) ? 0 : 2;

    // per-thread staging coordinates (16 B per lane per async op)
    const int arow = tid >> 3;                // 0..31
    const int ac4  = (tid & 7) << 2;          // 0,4,...,28

    v8f acc0 = {};   // columns c0 + 0..15
    v8f acc1 = {};   // columns c0 + 16..31

    // ---- prologue: stage chunk 0 into buffer 0
    async_cp16(adj + (size_t)arow * NN + c0 + ac4, &aS[0][arow][ac4]);
    #pragma unroll
    for (int it = 0; it < 4; ++it) {
        const int lin = tid + it * 256;       // 0..1023
        const int row = lin >> 5;
        const int h4  = (lin & 31) << 2;
        async_cp16(y + (size_t)row * HH + h4, &yS[0][row][h4]);
    }
    wait_async0();
    __syncthreads();

    int buf = 0;
    for (int kc = 0; kc < NN; kc += KC) {
        // ---- issue async stage of next chunk into the other buffer
        if (kc + KC < NN) {
            const int nb = buf ^ 1;
            async_cp16(adj + (size_t)(kc + KC + arow) * NN + c0 + ac4,
                       &aS[nb][arow][ac4]);
            #pragma unroll
            for (int it = 0; it < 4; ++it) {
                const int lin = tid + it * 256;
                const int row = lin >> 5;
                const int h4  = (lin & 31) << 2;
                async_cp16(y + (size_t)(kc + KC + row) * HH + h4,
                           &yS[nb][row][h4]);
            }
        }

        // ---- 8 WMMA K-steps of 4 over current chunk (overlaps async copies)
        #pragma unroll 4
        for (int k = 0; k < KC; k += 4) {
            v2f a, b0, b1;
            a.x  = yS[buf][k + kb][h0 + m];
            a.y  = yS[buf][k + kb + 1][h0 + m];
            b0.x = fmaxf(aS[buf][k + kb][m], 0.0f);        // edge mask
            b0.y = fmaxf(aS[buf][k + kb + 1][m], 0.0f);
            b1.x = fmaxf(aS[buf][k + kb][16 + m], 0.0f);
            b1.y = fmaxf(aS[buf][k + kb + 1][16 + m], 0.0f);
            acc0 = __builtin_amdgcn_wmma_f32_16x16x4_f32(
                       false, a, false, b0, (short)0, acc0, false, false);
            acc1 = __builtin_amdgcn_wmma_f32_16x16x4_f32(
                       false, a, false, b1, (short)0, acc1, false, false);
        }

        wait_async0();        // next chunk landed (latency hidden by compute)
        __syncthreads();
        buf ^= 1;
    }

    // ---- store D tiles: vgpr g holds M = g (lanes 0-15) / g+8 (lanes 16-31)
    const int rbase = h0 + ((lane < 16) ? 0 : 8);
    const int col0 = c0 + m;
    const int col1 = c0 + 16 + m;
    #pragma unroll
    for (int g = 0; g < 8; ++g) {
        zT[(size_t)(rbase + g) * NN + col0] = acc0[g];
        zT[(size_t)(rbase + g) * NN + col1] = acc1[g];
    }
}

// ---------------------------------------------------------------------------
// 7) fused epilogue per node c:
//    u = dinv[c]*(zT[:,c] + y[c,:]) + conv_b          (self-loop + norm + bias)
//    t = u @ mlp_w^T + mlp_b ;  out = relu(layernorm(t))
// ---------------------------------------------------------------------------
__global__ void k_epilogue(const float* __restrict__ zT, const float* __restrict__ y,
                           const float* __restrict__ dinv,
                           const float* __restrict__ conv_b,
                           const float* __restrict__ mlp_w, const float* __restrict__ mlp_b,
                           const float* __restrict__ ln_g, const float* __restrict__ ln_b,
                           float* __restrict__ xout) {
    __shared__ float u[HH];
    __shared__ float tb[HH];
    const int c = blockIdx.x;
    const int h = threadIdx.x;                 // 128 threads
    const float dv = dinv[c];
    u[h] = dv * (zT[(size_t)h * NN + c] + y[(size_t)c * HH + h]) + conv_b[h];
    __syncthreads();
    float t = mlp_b[h];
    const float* wr = mlp_w + (size_t)h * HH;
    #pragma unroll 8
    for (int k = 0; k < HH; ++k) t = fmaf(u[k], wr[k], t);
    tb[h] = t;
    __syncthreads();
    float s = 0.0f, s2 = 0.0f;
    #pragma unroll 8
    for (int k = 0; k < HH; ++k) { float v = tb[k]; s += v; s2 = fmaf(v, v, s2); }
    const float mu  = s * (1.0f / HH);
    const float var = s2 * (1.0f / HH) - mu * mu;
    float o = (t - mu) * __frsqrt_rn(var + LN_EPS) * ln_g[h] + ln_b[h];
    xout[(size_t)c * HH + h] = fmaxf(o, 0.0f);
}

// ---------------------------------------------------------------------------
extern "C" void kernel_launch(void* const* d_in, const int* in_sizes, int n_in,
                              void* d_out, int out_size, void* d_ws, size_t ws_size,
                              hipStream_t stream) {
    const float* X      = (const float*)d_in[0];
    const float* adj    = (const float*)d_in[1];
    const float* lin1_w = (const float*)d_in[2];
    const float* lin1_b = (const float*)d_in[3];
    const float* conv_w = (const float*)d_in[4];
    const float* conv_b = (const float*)d_in[5];
    const float* mlp_w  = (const float*)d_in[6];
    const float* mlp_b  = (const float*)d_in[7];
    const float* ln_g   = (const float*)d_in[8];
    const float* ln_b   = (const float*)d_in[9];

    float* out  = (float*)d_out;
    float* outA = out + (size_t)NN * HH;      // adj_A copy region

    float* ws   = (float*)d_ws;
    float* xa   = ws;                          // [N,H]
    float* xb   = xa + (size_t)NN * HH;        // [N,H]
    float* y    = xb + (size_t)NN * HH;        // [N,H]
    float* zT   = y  + (size_t)NN * HH;        // [H,N]
    float* deg  = zT + (size_t)NN * HH;        // [N]
    float* dinv = deg + NN;                    // [N]

    k_lin1    <<<NN,        128, 0, stream>>>(X, lin1_w, lin1_b, xa);
    k_deg_init<<<NN / 256,  256, 0, stream>>>(deg);
    k_deg_copy<<<512,       256, 0, stream>>>(adj, outA, deg);
    k_dinv    <<<NN / 256,  256, 0, stream>>>(deg, dinv);

    const float* xin = xa;
    for (int l = 0; l < 2; ++l) {
        float* xout = (l == 0) ? xb : out;    // final layer writes z_mu directly
        k_xw      <<<NN / 16, 128, 0, stream>>>(xin, conv_w + (size_t)l * HH * HH, dinv, y);
        k_spmm    <<<NN / CT, 256, 0, stream>>>(adj, y, zT);
        k_epilogue<<<NN,      128, 0, stream>>>(zT, y, dinv,
                                                conv_b + (size_t)l * HH,
                                                mlp_w  + (size_t)l * HH * HH,
                                                mlp_b  + (size_t)l * HH,
                                                ln_g   + (size_t)l * HH,
                                                ln_b   + (size_t)l * HH,
                                                xout);
        xin = xout;
    }
}